// MethodCPhysicsLoss_85555748537207
// MI455X (gfx1250) — compile-verified
//
#include <hip/hip_runtime.h>
#include <math.h>

typedef float v2f __attribute__((ext_vector_type(2)));
typedef float v8f __attribute__((ext_vector_type(8)));

// ---------------- wave / block reduction helpers (wave32) ----------------
__device__ __forceinline__ float waveSum(float v) {
#pragma unroll
  for (int o = 16; o > 0; o >>= 1) v += __shfl_xor(v, o, 32);
  return v;
}
__device__ __forceinline__ float waveMax(float v) {
#pragma unroll
  for (int o = 16; o > 0; o >>= 1) v = fmaxf(v, __shfl_xor(v, o, 32));
  return v;
}

template <int NV>
__device__ __forceinline__ void blockAtomicAdd(const float (&vals)[NV], float* acc, int accBase) {
  __shared__ float sm[NV][8];
  const int lane = threadIdx.x & 31;
  const int wid  = threadIdx.x >> 5;
#pragma unroll
  for (int i = 0; i < NV; i++) {
    float v = waveSum(vals[i]);
    if (lane == 0) sm[i][wid] = v;
  }
  __syncthreads();
  if (wid == 0) {
#pragma unroll
    for (int i = 0; i < NV; i++) {
      float v = (lane < (int)(blockDim.x >> 5)) ? sm[i][lane] : 0.f;
      v = waveSum(v);
      if (lane == 0) atomicAdd(&acc[accBase + i], v);
    }
  }
}

// d^order/dxi^order of xi^p evaluated at xi (matches reference _poly_basis)
__device__ __forceinline__ float dmono(int p, int order, float xi) {
  float c = 1.f;
  for (int r = 0; r < order; r++) c *= (float)(p - r);
  int e = p - order;
  if (e < 0 || c == 0.f) return 0.f;
  float x = 1.f;
  for (int i = 0; i < e; i++) x *= xi;
  return c * x;
}

// ---------------- workspace accumulator indices ----------------
// 0 sum(prop_E)  1 sum(prop_A)  2 sum(prop_I22)  3 sum(len)  4 max(|q|) (bits)
// 5 eqNum  6 freeCnt  7 freeNum  8 freeFaceCnt  9 s0  10 s1  11 s2  12 supD  13 supR
// 14 L_N raw  15 end-M raw  16 end-V raw  17 L_Mk raw  18 L_Mpp raw

__global__ void k_zero(float* p, long n) {
  long i = (long)blockIdx.x * blockDim.x + threadIdx.x;
  long stride = (long)gridDim.x * blockDim.x;
  for (; i < n; i += stride) p[i] = 0.f;
}

__global__ void k_props(const float* pE, const float* pA, const float* pI,
                        const float* len, const float* load, int E, float* acc) {
  int i = blockIdx.x * blockDim.x + threadIdx.x;
  float v[4] = {0.f, 0.f, 0.f, 0.f};
  float q = 0.f;
  if (i < E) {
    v[0] = pE[i]; v[1] = pA[i]; v[2] = pI[i]; v[3] = len[i];
    float a = load[3 * (size_t)i], b = load[3 * (size_t)i + 1], c = load[3 * (size_t)i + 2];
    q = __builtin_amdgcn_sqrtf(a * a + b * b + c * c);
  }
  blockAtomicAdd<4>(v, acc, 0);
  __shared__ float smx[8];
  const int lane = threadIdx.x & 31, wid = threadIdx.x >> 5;
  float m = waveMax(q);
  if (lane == 0) smx[wid] = m;
  __syncthreads();
  if (wid == 0) {
    float mm = (lane < (int)(blockDim.x >> 5)) ? smx[lane] : 0.f;
    mm = waveMax(mm);
    if (lane == 0) atomicMax((unsigned int*)&acc[4], __float_as_uint(mm));
  }
}

__global__ void k_scatter(const float* pred, const float* fmask,
                          const long long* feid, const long long* fisA,
                          int N, int E, float* ffA, float* ffB) {
  int n = blockIdx.x * blockDim.x + threadIdx.x;
  if (n >= N) return;
  const float* fr = pred + (size_t)n * 15 + 3;
#pragma unroll
  for (int f = 0; f < 4; f++) {
    if (fmask[(size_t)n * 4 + f] > 0.5f) {
      long long e = feid[(size_t)n * 4 + f];
      if (e >= 0 && e < (long long)E) {
        long long isA = fisA[(size_t)n * 4 + f];
        float* dst = (isA == 1) ? ffA : ((isA == 0) ? ffB : nullptr);
        if (dst) {
          dst[3 * e + 0] = fr[3 * f + 0];
          dst[3 * e + 1] = fr[3 * f + 1];
          dst[3 * e + 2] = fr[3 * f + 2];
        }
      }
    }
  }
}

__global__ void k_node(const float* pred, const float* F_ext, const float* bcd,
                       const float* bcr, const float* fmask, int N, float* acc) {
  int n = blockIdx.x * blockDim.x + threadIdx.x;
  float v[9] = {0.f, 0.f, 0.f, 0.f, 0.f, 0.f, 0.f, 0.f, 0.f};
  if (n < N) {
    const float* P = pred + (size_t)n * 15;
    float d0 = P[0], d1 = P[1], d2 = P[2];
    float s0 = 0.f, s1 = 0.f, s2 = 0.f, freeN = 0.f, ffc = 0.f;
#pragma unroll
    for (int f = 0; f < 4; f++) {
      float f0 = P[3 + 3 * f], f1 = P[4 + 3 * f], f2 = P[5 + 3 * f];
      s0 += f0; s1 += f1; s2 += f2;
      float fm = (fmask[(size_t)n * 4 + f] < 0.5f) ? 1.f : 0.f;
      freeN += fm * (f0 * f0 + f1 * f1 + f2 * f2);
      ffc += fm;
    }
    float r0 = s0 - F_ext[3 * (size_t)n];
    float r1 = s1 - F_ext[3 * (size_t)n + 1];
    float r2 = s2 - F_ext[3 * (size_t)n + 2];
    float bd = bcd[n], br = bcr[n];
    float fn = (bd < 0.5f) ? 1.f : 0.f;
    float sd = (bd > 0.5f) ? 1.f : 0.f;
    float sr = (br > 0.5f) ? 1.f : 0.f;
    v[0] = fn * (r0 * r0 + r1 * r1 + r2 * r2);
    v[1] = fn;
    v[2] = freeN;
    v[3] = ffc;
    v[4] = sd * d0 * d0;
    v[5] = sd * d1 * d1;
    v[6] = sr * d2 * d2;
    v[7] = sd;
    v[8] = sr;
  }
  blockAtomicAdd<9>(v, acc, 5);
}

// ---------------- WMMA polynomial kernel: 16 elements per wave ----------------
// D = A(16x4) x B(4x16) via V_WMMA_F32_16X16X4_F32.
// B columns 0-4: order-0 basis (M); 5-9: order-1 (dM, /L applied after);
// 10-14: order-2 (d2M, /L^2 after); col 15 zero. cw eval: two chained WMMAs (K=6).
// Endpoint values go to planar MV[4E]: [0..E) M_A, [E..2E) M_B,
// [2E..3E) V_A, [3E..4E) V_B — one predicated store per r.
__global__ void __launch_bounds__(256) k_epoly(
    const float* cM, const float* cw, const float* pE, const float* pI,
    const float* len, const float* load, int E, float* acc, float* MV) {
  const int lane = threadIdx.x & 31;
  const int wid  = threadIdx.x >> 5;
  const long base = ((long)blockIdx.x * (blockDim.x >> 5) + wid) * 16;
  if (base >= E) return;  // wave-uniform exit: EXEC stays all-ones for WMMA

  const int col  = lane & 15;
  const int half = lane >> 4;
  const int p0   = half * 2;

  // --- B operands (constant basis, per-lane per ISA B layout:
  //     lane -> column, VGPR0/1 -> K = {0,1} (lanes 0-15) or {2,3} (lanes 16-31))
  int kpt, order;
  if (col < 5)       { kpt = col;      order = 0; }
  else if (col < 10) { kpt = col - 5;  order = 1; }
  else if (col < 15) { kpt = col - 10; order = 2; }
  else               { kpt = 0;        order = 0; }
  const float xi = 0.25f * (float)kpt;
  v2f bM, bW1, bW2;
  if (col < 15) { bM.x = dmono(p0, order, xi); bM.y = dmono(p0 + 1, order, xi); }
  else          { bM.x = 0.f; bM.y = 0.f; }
  if (col < 5) {
    bW1.x = dmono(p0, 2, xi);  bW1.y = dmono(p0 + 1, 2, xi);
    bW2.x = (half == 0) ? dmono(4, 2, xi) : 0.f;   // global coeff p=4
    bW2.y = (half == 0) ? dmono(5, 2, xi) : 0.f;   // global coeff p=5
  } else { bW1.x = 0.f; bW1.y = 0.f; bW2.x = 0.f; bW2.y = 0.f; }

  // --- A operands (ISA 32-bit 16x4 A layout: lane -> row, half selects K pair)
  const int row = lane & 15;
  const long e  = base + row;
  const bool vr = (e < (long)E);
  const long ec = vr ? e : (long)E - 1;
  v2f aM, aW1, aW2;
  aM.x  = cM[ec * 4 + p0];     aM.y  = cM[ec * 4 + p0 + 1];
  aW1.x = cw[ec * 6 + p0];     aW1.y = cw[ec * 6 + p0 + 1];
  aW2.x = (half == 0) ? cw[ec * 6 + 4] : 0.f;
  aW2.y = (half == 0) ? cw[ec * 6 + 5] : 0.f;
  if (!vr) { aM.x = aM.y = aW1.x = aW1.y = aW2.x = aW2.y = 0.f; }

  // --- per-lane scalar cache for this wave's 16 elements (lane&15 -> row)
  const float Lsc    = len[ec];
  const float invLsc = __builtin_amdgcn_rcpf(Lsc);
  const float EIiL2  = pE[ec] * pI[ec] * invLsc * invLsc;  // EI / L^2
  const float q0l = load[ec * 3], q1l = load[ec * 3 + 1], q2l = load[ec * 3 + 2];
  const float qmsc = __builtin_amdgcn_sqrtf(q0l * q0l + q1l * q1l + q2l * q2l);

  if (base + 2048 < (long)E) {              // speculative prefetch of future A tiles
    __builtin_prefetch(&cM[(base + 2048) * 4], 0, 1);
    __builtin_prefetch(&cw[(base + 2048) * 6], 0, 1);
  }

  const v8f z = {0.f, 0.f, 0.f, 0.f, 0.f, 0.f, 0.f, 0.f};
  v8f DM = __builtin_amdgcn_wmma_f32_16x16x4_f32(false, aM,  false, bM,  (short)0, z,  false, false);
  v8f DW = __builtin_amdgcn_wmma_f32_16x16x4_f32(false, aW1, false, bW1, (short)0, z,  false, false);
  DW     = __builtin_amdgcn_wmma_f32_16x16x4_f32(false, aW2, false, bW2, (short)0, DW, false, false);

  // --- consume D in-place (VGPR r -> element base+r(+8), lane -> column)
  const float mkMask  = (col < 5) ? 1.f : 0.f;
  const float mppMask = (col >= 11 && col <= 13) ? 1.f : 0.f;
  // endpoint role: col 0 -> M_A, 4 -> M_B, 5 -> V_A, 9 -> V_B (one store per r)
  const bool storeAct = (col == 0) || (col == 4) || (col == 5) || (col == 9);
  const bool scaleByInvL = (col == 5) || (col == 9);
  const size_t kind = (col == 0) ? 0 : (col == 4) ? 1 : (col == 5) ? 2 : 3;
  float* MVk = MV + kind * (size_t)E;

  float mk = 0.f, mpp = 0.f;
#pragma unroll
  for (int r = 0; r < 8; r++) {
    const int srcl = r + (half << 3);
    const float invLr = __shfl(invLsc, srcl, 32);
    const float eiL2r = __shfl(EIiL2,  srcl, 32);
    const float qmr   = __shfl(qmsc,   srcl, 32);
    const long em = base + r + (half << 3);
    const float vm = (em < (long)E) ? 1.f : 0.f;
    const float invL2 = invLr * invLr;
    const float d = DM[r];
    const float t1 = (d - eiL2r * DW[r]) * (mkMask * vm);   // M - EI*d2w_dx2
    mk += t1 * t1;
    const float t2 = (d * invL2 + qmr) * (mppMask * vm);    // d2M_dx2 + q
    mpp += t2 * t2;
    if (storeAct && em < (long)E) {
      MVk[em] = scaleByInvL ? d * invLr : d;
    }
  }
  mk = waveSum(mk);
  mpp = waveSum(mpp);
  if (lane == 0) { atomicAdd(&acc[17], mk); atomicAdd(&acc[18], mpp); }
}

__global__ void k_emain(const float* pred, const long long* conn, const float* dirs,
                        const float* pE, const float* pA, const float* len,
                        const float* ffA, const float* ffB, const float* MV,
                        int E, float* acc) {
  int e = blockIdx.x * blockDim.x + threadIdx.x;
  float vals[3] = {0.f, 0.f, 0.f};
  if (e < E) {
    const long a = (long)conn[2 * (size_t)e];
    const long b = (long)conn[2 * (size_t)e + 1];
    const float* PA = pred + a * 15;
    const float* PB = pred + b * 15;
    const float c = dirs[3 * (size_t)e], s = dirs[3 * (size_t)e + 2];
    const float dA0 = PA[0] * c + PA[1] * s;
    const float dB0 = PB[0] * c + PB[1] * s;
    const float fAx = ffA[3 * (size_t)e], fAy = ffA[3 * (size_t)e + 1], fAz = ffA[3 * (size_t)e + 2];
    const float fBx = ffB[3 * (size_t)e], fBy = ffB[3 * (size_t)e + 1], fBz = ffB[3 * (size_t)e + 2];
    const float fa0 =  fAx * c + fAy * s, fa1 = -fAx * s + fAy * c, fa2 = fAz;
    const float fb0 =  fBx * c + fBy * s, fb1 = -fBx * s + fBy * c, fb2 = fBz;
    const float Nax = pE[e] * pA[e] * (dB0 - dA0) * __builtin_amdgcn_rcpf(len[e]);
    const float t1 = fa0 + Nax, t2 = fb0 - Nax;
    vals[0] = t1 * t1 + t2 * t2;                          // L_N numerator
    const float m1 = fa2 + MV[e];                         // + M_A
    const float m2 = fb2 - MV[(size_t)E + e];             // - M_B
    vals[1] = m1 * m1 + m2 * m2;                          // L_end moment part
    const float v1 = fa1 + MV[2 * (size_t)E + e];         // + V_A
    const float v2 = fb1 - MV[3 * (size_t)E + e];         // - V_B
    vals[2] = v1 * v1 + v2 * v2;                          // L_end shear part
  }
  blockAtomicAdd<3>(vals, acc, 14);
}

__global__ void k_final(const float* acc, int E, float* out) {
  if (threadIdx.x != 0 || blockIdx.x != 0) return;
  const float fE = (float)E;
  const float Er = acc[0] / fE, Ar = acc[1] / fE, Ir = acc[2] / fE, Lr = acc[3] / fE;
  float q0 = acc[4];                       // bit-pattern max of non-negative floats
  q0 = (q0 < 1e-10f) ? 1.f : q0;
  const float Nref = fmaxf(Er * Ar * 0.001f / Lr, 1e-6f);
  const float Mref = fmaxf(Er * Ir * 0.001f / (Lr * Lr), 1e-6f);
  const float Vref = fmaxf(Mref / Lr, 1e-6f);
  const float qref = fmaxf(q0, 1e-6f);
  const float Fref = fmaxf(q0 * Lr, 1e-6f);
  const float iF2 = 1.f / (Fref * Fref);
  const float L_eq   = acc[5]  * iF2 / fmaxf(acc[6], 1.f);
  const float L_free = acc[7]  * iF2 / fmaxf(acc[8] * 3.f, 1.f);
  const float L_sup  = acc[9]  / fmaxf(acc[12], 1.f)
                     + acc[10] / fmaxf(acc[12], 1.f)
                     + acc[11] / fmaxf(acc[13], 1.f);
  const float L_N    = acc[14] / (Nref * Nref) / fE;
  const float L_end  = acc[15] / (Mref * Mref) / fE + acc[16] / (Vref * Vref) / fE;
  const float L_Mk   = acc[17] / (Mref * Mref) / (fE * 5.f);
  const float L_Mpp  = acc[18] / (qref * qref) / (fE * 3.f);
  out[0] = L_eq + L_free + L_sup + L_N + L_Mk + L_Mpp + L_end;
}

extern "C" void kernel_launch(void* const* d_in, const int* in_sizes, int n_in,
                              void* d_out, int out_size, void* d_ws, size_t ws_size,
                              hipStream_t stream) {
  const float* pred  = (const float*)d_in[0];
  const float* F_ext = (const float*)d_in[1];
  const float* bcd   = (const float*)d_in[2];
  const float* bcr   = (const float*)d_in[3];
  const float* fmask = (const float*)d_in[4];
  const float* dirs  = (const float*)d_in[5];
  const float* pE    = (const float*)d_in[6];
  const float* pA    = (const float*)d_in[7];
  const float* pI    = (const float*)d_in[8];
  const float* len   = (const float*)d_in[9];
  const float* load  = (const float*)d_in[10];
  const float* cw    = (const float*)d_in[11];
  const float* cM    = (const float*)d_in[12];
  const long long* conn = (const long long*)d_in[13];
  const long long* feid = (const long long*)d_in[14];
  const long long* fisA = (const long long*)d_in[15];

  const int N = in_sizes[2];      // bc_disp has N entries
  const int E = in_sizes[6];      // prop_E has E entries

  float* W   = (float*)d_ws;
  float* acc = W;                              // 32 floats
  float* ffA = W + 32;                         // 3E
  float* ffB = ffA + (size_t)3 * E;            // 3E
  float* MV  = ffB + (size_t)3 * E;            // 4E (M_A, M_B, V_A, V_B)

  const int B = 256;
  const long zn = 32 + (long)6 * E;            // acc + ffA + ffB (MV fully overwritten)
  k_zero<<<4096, B, 0, stream>>>(W, zn);
  k_props<<<(E + B - 1) / B, B, 0, stream>>>(pE, pA, pI, len, load, E, acc);
  k_scatter<<<(N + B - 1) / B, B, 0, stream>>>(pred, fmask, feid, fisA, N, E, ffA, ffB);
  k_node<<<(N + B - 1) / B, B, 0, stream>>>(pred, F_ext, bcd, bcr, fmask, N, acc);
  const long waves  = ((long)E + 15) / 16;
  const long blocks = (waves + 7) / 8;         // 8 waves per 256-thread block
  k_epoly<<<(int)blocks, B, 0, stream>>>(cM, cw, pE, pI, len, load, E, acc, MV);
  k_emain<<<(E + B - 1) / B, B, 0, stream>>>(pred, conn, dirs, pE, pA, len, ffA, ffB,
                                             MV, E, acc);
  k_final<<<1, 64, 0, stream>>>(acc, E, (float*)d_out);
}